// GeneralLVeG_9698036154936
// MI455X (gfx1250) — compile-verified
//
#include <hip/hip_runtime.h>
#include <cmath>

// ---------------- problem constants ----------------
#define BB     16
#define TT     256
#define DD_DIM 512
#define LL_    51
#define L2     2601          // L*L
#define BLL    41616         // B*L*L
#define BL     816           // B*L
#define TOT    10653696      // T*B*L*L
#define NCOLS  18360         // 3L + L^2 + 2L^2 + 4L^2
#define NP     18432         // NCOLS padded to multiple of 128

#define LOG2PI_F 1.8378770664093453f

// column block offsets inside the fused GEMM
#define OFF_SW    0
#define OFF_SMU   51
#define OFF_SVAR  102
#define OFF_TW    153
#define OFF_TMU   2754
#define OFF_TRAW  7956

typedef float v2f __attribute__((ext_vector_type(2)));
typedef float v8f __attribute__((ext_vector_type(8)));

__device__ __forceinline__ float clip1(float v) {
    return fminf(1.0f, fmaxf(-1.0f, v));
}

// ---------------- kernel 0a: pack weights into one (D, NP) row-major buffer ----------------
__global__ __launch_bounds__(256) void pack_w_kernel(
    const float* __restrict__ Ws_w, const float* __restrict__ Ws_mu, const float* __restrict__ Ws_var,
    const float* __restrict__ Wt_w, const float* __restrict__ Wt_mu, const float* __restrict__ Wt_var,
    float* __restrict__ Wp)
{
    const int idx = blockIdx.x * 256 + threadIdx.x;        // D*NP = 9437184 total
    if (idx >= DD_DIM * NP) return;
    const int k = idx / NP;
    const int c = idx % NP;
    float v = 0.0f;
    if      (c < OFF_SMU)  v = Ws_w [k * LL_   + (c - OFF_SW)];
    else if (c < OFF_SVAR) v = Ws_mu[k * LL_   + (c - OFF_SMU)];
    else if (c < OFF_TW)   v = Ws_var[k * LL_  + (c - OFF_SVAR)];
    else if (c < OFF_TMU)  v = Wt_w [k * L2    + (c - OFF_TW)];
    else if (c < OFF_TRAW) v = Wt_mu[k * (2*L2)+ (c - OFF_TMU)];
    else if (c < NCOLS)    v = Wt_var[k * (4*L2)+ (c - OFF_TRAW)];
    Wp[idx] = v;
}

__global__ __launch_bounds__(256) void pack_b_kernel(
    const float* __restrict__ bs_w, const float* __restrict__ bs_mu, const float* __restrict__ bs_var,
    const float* __restrict__ bt_w, const float* __restrict__ bt_mu, const float* __restrict__ bt_var,
    float* __restrict__ bp)
{
    const int c = blockIdx.x * 256 + threadIdx.x;
    if (c >= NP) return;
    float v = 0.0f;
    if      (c < OFF_SMU)  v = bs_w [c - OFF_SW];
    else if (c < OFF_SVAR) v = bs_mu[c - OFF_SMU];
    else if (c < OFF_TW)   v = bs_var[c - OFF_SVAR];
    else if (c < OFF_TMU)  v = bt_w [c - OFF_TW];
    else if (c < OFF_TRAW) v = bt_mu[c - OFF_TMU];
    else if (c < NCOLS)    v = bt_var[c - OFF_TRAW];
    bp[c] = v;
}

// ---------------- kernel 1: fp32 WMMA GEMM  Y(4096,NP) = X(4096,512) * W(512,NP) + bias ----------------
// Tile: 64(M) x 128(N) per 256-thread block; 8 waves, each owning a 16x64 strip.
// LDS layouts are "fragment-ready":
//   A: pitch 18 (even -> aligned b64 frag loads; 18*ml mod 64 distinct -> conflict-free)
//   B: K-pair interleaved [kpair][2*n+parity], pitch 288 (mod 64 == 32 -> lane halves disjoint)
// Double-buffered LDS, one barrier per K-chunk; next chunk staged in registers during WMMA.
#define BM 64
#define BN 128
#define KC 16
#define AS_PITCH 18
#define BP_PITCH 288          // 2*BN + 32

__global__ __launch_bounds__(256) void gemm_kernel(
    const float* __restrict__ X, const float* __restrict__ W,
    const float* __restrict__ bias, float* __restrict__ Y)
{
    __shared__ float As[2][BM * AS_PITCH];          // 2 * 1152 floats
    __shared__ float Bs[2][(KC / 2) * BP_PITCH];    // 2 * 2304 floats

    const int tid  = threadIdx.x;
    const int lane = tid & 31;
    const int wave = tid >> 5;
    const int half = lane >> 4;         // K-pair selector per ISA 16x4 layout
    const int ml   = lane & 15;
    const int m0   = (wave & 3) * 16;   // wave's M subtile
    const int n0   = (wave >> 2) * 64;  // wave's 64-wide N strip (4 subtiles)
    const int rowBase = blockIdx.y * BM;
    const int colBase = blockIdx.x * BN;

    v8f acc[4];
    #pragma unroll
    for (int s = 0; s < 4; ++s)
        #pragma unroll
        for (int r = 0; r < 8; ++r) acc[s][r] = 0.0f;

    // global->LDS staging assignments
    const int am = tid >> 2, akq = tid & 3;    // A: 64 rows x 4 float4
    const int br = tid >> 5, bnq = tid & 31;   // B: row pair (2br, 2br+1), 32 float4 cols

    const float* __restrict__ Aptr  = X + (size_t)(rowBase + am) * DD_DIM + akq * 4;
    const float* __restrict__ B0ptr = W + (size_t)(2 * br) * NP + colBase + bnq * 4;
    const float* __restrict__ B1ptr = B0ptr + NP;

    // ---- prologue: chunk 0 into buffer 0 ----
    float4 aR  = *(const float4*)(Aptr);
    float4 b0R = *(const float4*)(B0ptr);
    float4 b1R = *(const float4*)(B1ptr);
    {
        float* ad = &As[0][am * AS_PITCH + akq * 4];
        v2f t0; t0.x = aR.x; t0.y = aR.y; *(v2f*)(ad)     = t0;
        v2f t1; t1.x = aR.z; t1.y = aR.w; *(v2f*)(ad + 2) = t1;
        float* bd = &Bs[0][br * BP_PITCH + bnq * 8];
        v2f p0; p0.x = b0R.x; p0.y = b1R.x; *(v2f*)(bd)     = p0;
        v2f p1; p1.x = b0R.y; p1.y = b1R.y; *(v2f*)(bd + 2) = p1;
        v2f p2; p2.x = b0R.z; p2.y = b1R.z; *(v2f*)(bd + 4) = p2;
        v2f p3; p3.x = b0R.w; p3.y = b1R.w; *(v2f*)(bd + 6) = p3;
    }

    int buf = 0;
    for (int kb = 0; kb < DD_DIM; kb += KC) {
        __syncthreads();   // buffer `buf` ready; all waves done with buffer buf^1
        const bool more = (kb + KC) < DD_DIM;
        float4 aN, b0N, b1N;
        if (more) {        // stage next chunk in registers while WMMAs run
            aN  = *(const float4*)(Aptr  + (kb + KC));
            b0N = *(const float4*)(B0ptr + (size_t)(kb + KC) * NP);
            b1N = *(const float4*)(B1ptr + (size_t)(kb + KC) * NP);
        }
        if (kb + 2 * KC < DD_DIM) {   // warm L2 for the chunk after next (global_prefetch_b8)
            __builtin_prefetch(Aptr  + (kb + 2 * KC), 0, 1);
            __builtin_prefetch(B0ptr + (size_t)(kb + 2 * KC) * NP, 0, 1);
        }
        #pragma unroll
        for (int k0 = 0; k0 < KC; k0 += 4) {
            // A fragment: lane<16 holds (k0,k0+1), lane>=16 holds (k0+2,k0+3) -- one aligned b64
            v2f a = *(const v2f*)&As[buf][(m0 + ml) * AS_PITCH + k0 + 2 * half];
            #pragma unroll
            for (int s = 0; s < 4; ++s) {
                const int nn = n0 + s * 16 + ml;
                v2f b = *(const v2f*)&Bs[buf][((k0 >> 1) + half) * BP_PITCH + nn * 2];
                acc[s] = __builtin_amdgcn_wmma_f32_16x16x4_f32(
                    false, a, false, b, (short)0, acc[s], false, false);
            }
        }
        if (more) {        // fill the other buffer (nobody reads it this iteration)
            const int nb = buf ^ 1;
            float* ad = &As[nb][am * AS_PITCH + akq * 4];
            v2f t0; t0.x = aN.x; t0.y = aN.y; *(v2f*)(ad)     = t0;
            v2f t1; t1.x = aN.z; t1.y = aN.w; *(v2f*)(ad + 2) = t1;
            float* bd = &Bs[nb][br * BP_PITCH + bnq * 8];
            v2f p0; p0.x = b0N.x; p0.y = b1N.x; *(v2f*)(bd)     = p0;
            v2f p1; p1.x = b0N.y; p1.y = b1N.y; *(v2f*)(bd + 2) = p1;
            v2f p2; p2.x = b0N.z; p2.y = b1N.z; *(v2f*)(bd + 4) = p2;
            v2f p3; p3.x = b0N.w; p3.y = b1N.w; *(v2f*)(bd + 6) = p3;
        }
        buf ^= 1;
    }

    // epilogue: C layout -> VGPR r: M = r + 8*half, N = ml
    #pragma unroll
    for (int s = 0; s < 4; ++s) {
        const int col = colBase + n0 + s * 16 + ml;
        const float bv = bias[col];
        #pragma unroll
        for (int r = 0; r < 8; ++r) {
            const int row = rowBase + m0 + r + 8 * half;
            Y[(size_t)row * NP + col] = acc[s][r] + bv;
        }
    }
}

// ---------------- kernel 2: per-(t,b,i,j) 2x2-Gaussian precomputation ----------------
__global__ __launch_bounds__(256) void transform_kernel(
    const float* __restrict__ Y,
    float* __restrict__ Z1, float* __restrict__ Aa, float* __restrict__ Bo,
    float* __restrict__ Dd, float* __restrict__ E0, float* __restrict__ E1)
{
    const int f = blockIdx.x * 256 + threadIdx.x;
    if (f >= TOT) return;
    const int t   = f / BLL;
    const int bij = f % BLL;
    const int b   = bij / L2;
    const int ij  = bij % L2;
    const float* __restrict__ row = Y + (size_t)(b * TT + t) * NP;

    const float2 mv = *(const float2*)(row + OFF_TMU  + ij * 2);   // 8B aligned (2754 even)
    const float4 rv = *(const float4*)(row + OFF_TRAW + ij * 4);   // 16B aligned (7956 % 4 == 0)

    const float d0  = expf(clip1(rv.x));
    const float d1  = expf(clip1(rv.w));
    const float off = 0.9f * tanhf(0.5f * (rv.y + rv.z)) * sqrtf(d0 * d1);

    const float det = d0 * d1 - off * off;
    const float inv = 1.0f / det;
    const float a   = d1 * inv;
    const float bo  = -off * inv;
    const float dd  = d0 * inv;
    const float e0  = a * mv.x + bo * mv.y;
    const float e1  = bo * mv.x + dd * mv.y;
    const float quad = e0 * e0 * d0 + 2.0f * e0 * e1 * off + e1 * e1 * d1;
    const float z1  = -0.5f * (2.0f * LOG2PI_F + logf(det) + quad);

    Z1[f] = z1; Aa[f] = a; Bo[f] = bo; Dd[f] = dd; E0[f] = e0; E1[f] = e1;
}

// ---------------- kernel 3: s_var = exp(clip(.)) in time-major (T,B,L) ----------------
__global__ __launch_bounds__(256) void svar_kernel(
    const float* __restrict__ Y, float* __restrict__ SV)
{
    const int f = blockIdx.x * 256 + threadIdx.x;     // T*B*L = 208896
    if (f >= TT * BL) return;
    const int t = f / BL;
    const int r = f % BL;
    const int b = r / LL_;
    const int l = r % LL_;
    SV[f] = expf(clip1(Y[(size_t)(b * TT + t) * NP + OFF_SVAR + l]));
}

// ---------------- kernel 4: sequential scan over T ----------------
struct MergeOut { float scale, mu, var; };

__device__ __forceinline__ MergeOut merge_child(
    float z1, float a, float bo, float dd, float e0, float e1, float nmu, float nvar)
{
    const float l2   = 1.0f / nvar;
    const float eta2 = nmu * l2;
    const float zeta2 = -0.5f * (LOG2PI_F + logf(nvar) + nmu * nmu * l2);
    const float den = dd + l2;
    const float es  = e1 + eta2;
    const float la  = a - bo * bo / den;
    const float eta = e0 - (bo / den) * es;
    const float var = 1.0f / la;
    const float mu  = var * eta;
    const float zm  = -0.5f * (LOG2PI_F - logf(den) + es * es / den);
    const float zn  = -0.5f * (LOG2PI_F - logf(la) + eta * eta * var);
    return { z1 + zeta2 - zm - zn, mu, var };
}

__device__ __forceinline__ MergeOut merge_parent(
    float z1, float a, float bo, float dd, float e0, float e1, float nmu, float nvar)
{
    const float l2   = 1.0f / nvar;
    const float eta2 = nmu * l2;
    const float zeta2 = -0.5f * (LOG2PI_F + logf(nvar) + nmu * nmu * l2);
    const float den = a + l2;
    const float es  = e0 + eta2;
    const float la  = dd - bo * bo / den;
    const float eta = e1 - (bo / den) * es;
    const float var = 1.0f / la;
    const float mu  = var * eta;
    const float zm  = -0.5f * (LOG2PI_F - logf(den) + es * es / den);
    const float zn  = -0.5f * (LOG2PI_F - logf(la) + eta * eta * var);
    return { z1 + zeta2 - zm - zn, mu, var };
}

__global__ __launch_bounds__(256) void scan_kernel(
    const float* __restrict__ Y,
    const float* __restrict__ Z1, const float* __restrict__ Aa, const float* __restrict__ Bo,
    const float* __restrict__ Dd, const float* __restrict__ E0, const float* __restrict__ E1,
    const float* __restrict__ SV, float* __restrict__ out)
{
    const int bij = blockIdx.x * 256 + threadIdx.x;
    if (bij >= BLL) return;
    const int b  = bij / L2;
    const int ij = bij % L2;
    const int j  = ij % LL_;

    // ---- t = 0: child merge with s(0), then parent merge with transition t=1 ----
    const float* __restrict__ row0 = Y + (size_t)(b * TT) * NP;
    const float sw0  = row0[OFF_SW  + j];
    const float smu0 = row0[OFF_SMU + j];
    const float sv0  = SV[b * LL_ + j];
    const float tw0  = row0[OFF_TW + ij];

    MergeOut c0 = merge_child(Z1[bij], Aa[bij], Bo[bij], Dd[bij], E0[bij], E1[bij], smu0, sv0);
    const int f1 = BLL + bij;   // precomp at t=1
    MergeOut p0 = merge_parent(Z1[f1], Aa[f1], Bo[f1], Dd[f1], E0[f1], E1[f1], c0.mu, c0.var);

    out[(size_t)(b * TT) * L2 + ij] = c0.scale + p0.scale + tw0 + sw0;

    float mu_p  = p0.mu;
    float var_p = p0.var;

    // ---- t = 1..T-1 ----
    for (int t = 1; t < TT; ++t) {
        const float* __restrict__ rowt = Y + (size_t)(b * TT + t) * NP;
        const float smu = rowt[OFF_SMU + j];
        const float sw  = rowt[OFF_SW  + j];
        const float tw  = rowt[OFF_TW + ij];
        const float sv  = SV[t * BL + b * LL_ + j];

        const float vsum = var_p + sv;
        const float logv = logf(vsum);
        const float dm   = mu_p - smu;
        const float scale_c = -0.5f * (LOG2PI_F + logv + dm * dm / vsum);
        const float mu_c  = (mu_p * sv + smu * var_p) / vsum;
        const float var_c = vsum - 0.5f * logv;

        const size_t f = (size_t)t * BLL + bij;
        MergeOut pp = merge_parent(Z1[f], Aa[f], Bo[f], Dd[f], E0[f], E1[f], mu_c, var_c);
        mu_p  = pp.mu;
        var_p = pp.var;

        out[(size_t)(b * TT + t) * L2 + ij] = scale_c + pp.scale + tw + sw;
    }
}

// ---------------- launch ----------------
extern "C" void kernel_launch(void* const* d_in, const int* in_sizes, int n_in,
                              void* d_out, int out_size, void* d_ws, size_t ws_size,
                              hipStream_t stream) {
    const float* x      = (const float*)d_in[0];
    const float* Ws_w   = (const float*)d_in[1];
    const float* bs_w   = (const float*)d_in[2];
    const float* Ws_mu  = (const float*)d_in[3];
    const float* bs_mu  = (const float*)d_in[4];
    const float* Ws_var = (const float*)d_in[5];
    const float* bs_var = (const float*)d_in[6];
    const float* Wt_w   = (const float*)d_in[7];
    const float* bt_w   = (const float*)d_in[8];
    const float* Wt_mu  = (const float*)d_in[9];
    const float* bt_mu  = (const float*)d_in[10];
    const float* Wt_var = (const float*)d_in[11];
    const float* bt_var = (const float*)d_in[12];
    float* out = (float*)d_out;

    // workspace layout (floats)
    float* ws   = (float*)d_ws;
    float* Wp   = ws;                               // D*NP      = 9,437,184
    float* bp   = Wp + (size_t)DD_DIM * NP;         // NP        = 18,432
    float* Y    = bp + NP;                          // 4096*NP   = 75,497,472
    float* Z1   = Y  + (size_t)(BB * TT) * NP;      // TOT each
    float* Aa   = Z1 + (size_t)TOT;
    float* Bo   = Aa + (size_t)TOT;
    float* Dd   = Bo + (size_t)TOT;
    float* E0   = Dd + (size_t)TOT;
    float* E1   = E0 + (size_t)TOT;
    float* SV   = E1 + (size_t)TOT;                 // T*B*L = 208,896
    (void)ws_size; (void)in_sizes; (void)n_in; (void)out_size;

    // 0) pack weights & biases
    pack_w_kernel<<<(DD_DIM * NP + 255) / 256, 256, 0, stream>>>(
        Ws_w, Ws_mu, Ws_var, Wt_w, Wt_mu, Wt_var, Wp);
    pack_b_kernel<<<(NP + 255) / 256, 256, 0, stream>>>(
        bs_w, bs_mu, bs_var, bt_w, bt_mu, bt_var, bp);

    // 1) fused GEMM (fp32 WMMA): Y = x @ [all weights] + bias
    dim3 ggrid(NP / BN, (BB * TT) / BM);            // 144 x 64
    gemm_kernel<<<ggrid, 256, 0, stream>>>(x, Wp, bp, Y);

    // 2) carry-independent 2x2-Gaussian precomputation
    transform_kernel<<<(TOT + 255) / 256, 256, 0, stream>>>(Y, Z1, Aa, Bo, Dd, E0, E1);

    // 3) s_var
    svar_kernel<<<(TT * BL + 255) / 256, 256, 0, stream>>>(Y, SV);

    // 4) sequential scan over T, parallel over B*L*L
    scan_kernel<<<(BLL + 255) / 256, 256, 0, stream>>>(Y, Z1, Aa, Bo, Dd, E0, E1, SV, out);
}